// PointRend_49709951484601
// MI455X (gfx1250) — compile-verified
//
#include <hip/hip_runtime.h>

#define B_    8
#define C_    256
#define H_    256
#define W_    256
#define HW_   65536
#define N_    4096
#define TOPK_ 4096

// d_out layout (flat concatenation, reference return order):
//   local_feats (B,C,N)  : 8*256*4096 = 8388608
//   sampled     (B,N,C)  : 8*4096*256 = 8388608
//   coords      (B,N,2)  :              65536
//   idx (float) (B,N)    :              32768
#define OFF_LOCAL  0ull
#define OFF_SAMP   8388608ull
#define OFF_COORDS 16777216ull
#define OFF_IDX    16842752ull

#define SELCAP 8192
#define PLANE_NT 512

#if defined(__has_builtin)
#if __has_builtin(__builtin_amdgcn_tensor_load_to_lds) && \
    __has_builtin(__builtin_amdgcn_s_wait_tensorcnt)
#define USE_TDM 1
#endif
#if __has_builtin(__builtin_amdgcn_global_load_async_to_lds_b128) && \
    __has_builtin(__builtin_amdgcn_s_wait_asynccnt)
#define USE_ASYNC_LDS 1
#endif
#endif

// Generic (GCC-style) vector types matching clang's builtin prototypes.
typedef int          v4i_ __attribute__((vector_size(16)));
typedef unsigned int v4u_ __attribute__((vector_size(16)));
typedef int          v8i_ __attribute__((vector_size(32)));
typedef __attribute__((address_space(1))) v4i_* gas_v4i;
typedef __attribute__((address_space(3))) v4i_* las_v4i;

// Single dynamic-LDS symbol shared by both kernels (avoid conflicting extern types).
extern __shared__ __align__(16) unsigned char dynsmem[];

// Monotonic float -> uint mapping (larger float => larger uint).
__device__ __forceinline__ unsigned f2ord(float f) {
  unsigned u = __float_as_uint(f);
  return (u & 0x80000000u) ? ~u : (u | 0x80000000u);
}

__device__ __forceinline__ unsigned keyAt(const float* __restrict__ pred,
                                          const int* __restrict__ edge, int i) {
  float unc = -fabsf(pred[i]);
  float m = (edge[i] == 1) ? unc : -1e30f;
  return f2ord(m);
}

// ---------------------------------------------------------------------------
// Kernel A: per-batch exact top-k via radix-select + in-LDS bitonic sort.
// One block per batch, 1024 threads. Dynamic LDS: sel[8192] u64 + hist[256] + scalars.
// ---------------------------------------------------------------------------
extern "C" __global__ void __launch_bounds__(1024)
topk_kernel(const float* __restrict__ pred_mask,
            const int* __restrict__ edge_masks,
            float* __restrict__ out) {
  const int b   = blockIdx.x;
  const int tid = threadIdx.x;
  const int nt  = blockDim.x;
  const float* pred = pred_mask + (size_t)b * HW_;
  const int*   edge = edge_masks + (size_t)b * HW_;

  unsigned long long* sel = (unsigned long long*)dynsmem;           // 8192*8 B
  unsigned* hist = (unsigned*)(dynsmem + (size_t)SELCAP * 8);       // 256*4 B
  unsigned* scal = (unsigned*)(dynsmem + (size_t)SELCAP * 8 + 1024);

  // ---- 4-pass radix select (MSB first) to find exact 32-bit threshold ----
  unsigned prefix = 0, kneed = TOPK_;
  for (int pass = 0; pass < 4; ++pass) {
    const int shift = 24 - 8 * pass;
    for (int i = tid; i < 256; i += nt) hist[i] = 0u;
    __syncthreads();
    for (int i = tid; i < HW_; i += nt) {
      unsigned u = keyAt(pred, edge, i);
      if (pass == 0 || (u >> (shift + 8)) == prefix)
        atomicAdd(&hist[(u >> shift) & 255u], 1u);
    }
    __syncthreads();
    if (tid == 0) {
      unsigned acc = 0; int bsel = 0;
      for (int bin = 255; bin >= 0; --bin) {
        unsigned cct = hist[bin];
        if (acc + cct >= kneed) { bsel = bin; break; }
        acc += cct;
      }
      scal[0] = (prefix << 8) | (unsigned)bsel;
      scal[1] = kneed - acc;  // still needed inside selected bin
    }
    __syncthreads();
    prefix = scal[0];
    kneed  = scal[1];
    __syncthreads();
  }
  const unsigned T = prefix;  // exact threshold key

  // ---- compact all candidates (key >= T) into LDS as sortable u64 keys ----
  if (tid == 0) scal[2] = 0u;
  __syncthreads();
  for (int i = tid; i < HW_; i += nt) {
    unsigned u = keyAt(pred, edge, i);
    if (u >= T) {
      unsigned pos = atomicAdd(&scal[2], 1u);
      if (pos < SELCAP)
        sel[pos] = ((unsigned long long)u << 32) |
                   (unsigned long long)(0xFFFFFFFFu - (unsigned)i);  // idx asc on ties
    }
  }
  __syncthreads();
  unsigned total = scal[2]; if (total > SELCAP) total = SELCAP;
  for (int i = tid; i < SELCAP; i += nt)
    if ((unsigned)i >= total) sel[i] = 0ull;  // pad sinks to the end
  __syncthreads();

  // ---- bitonic sort, descending ----
  for (int kk = 2; kk <= SELCAP; kk <<= 1) {
    for (int j = kk >> 1; j > 0; j >>= 1) {
      for (int i = tid; i < SELCAP; i += nt) {
        int ixj = i ^ j;
        if (ixj > i) {
          unsigned long long a = sel[i], c = sel[ixj];
          bool up = ((i & kk) == 0);
          if (up ? (a < c) : (a > c)) { sel[i] = c; sel[ixj] = a; }
        }
      }
      __syncthreads();
    }
  }

  // ---- emit coords + idx (sorted: value desc, index asc) ----
  float* out_coords = out + OFF_COORDS;
  float* out_idx    = out + OFF_IDX;
  for (int j = tid; j < TOPK_; j += nt) {
    unsigned long long kk = sel[j];
    unsigned idx = 0xFFFFFFFFu - (unsigned)(kk & 0xFFFFFFFFull);
    float yi = (float)(idx / W_);
    float xi = (float)(idx % W_);
    size_t o = (size_t)b * TOPK_ + (size_t)j;
    out_coords[o * 2 + 0] = yi / (float)(H_ - 1);
    out_coords[o * 2 + 1] = xi / (float)(W_ - 1);
    out_idx[o] = (float)idx;
  }
}

// ---------------------------------------------------------------------------
// Kernel B: one block per (b,c) plane. Stage the full 256KB plane in LDS with
// the Tensor Data Mover (one DMA per plane, TENSORcnt), falling back to async
// global->LDS copies (ASYNCcnt), then do all bilinear samples and all 3x3
// local-feature means from LDS.
// ---------------------------------------------------------------------------
extern "C" __global__ void __launch_bounds__(PLANE_NT)
plane_kernel(const float* __restrict__ feature_map,
             const float* __restrict__ point_coords,
             float* __restrict__ out) {
  const int c   = blockIdx.x;
  const int b   = blockIdx.y;
  const int tid = threadIdx.x;
  const int nt  = blockDim.x;

  float* plane = (float*)dynsmem;  // 65536 floats = 256 KB of the 320 KB LDS
  const float* src = feature_map + ((size_t)b * C_ + (size_t)c) * HW_;

#if defined(USE_TDM)
  // One TDM DMA moves the whole 256x256 f32 tile. TDM ignores EXEC but issues
  // per wave, so only wave 0 launches it (16 issues would be 16 full DMAs).
  if (tid < 32) {
    unsigned lds_addr = (unsigned)(size_t)(__attribute__((address_space(3))) float*)plane;
    unsigned long long ga = (unsigned long long)(size_t)src;
    // D# group0: count=1 | lds_addr | global_addr[31:0] | global_addr[56:32], type=2
    v4u_ g0 = { 1u,
                lds_addr,
                (unsigned)(ga & 0xFFFFFFFFull),
                (unsigned)((ga >> 32) & 0x01FFFFFFull) | 0x80000000u };
    // D# group1: data_size=4B; tensor_dim0=256, tensor_dim1=256;
    //            tile_dim0=256, tile_dim1=256, tile_dim2 unused;
    //            tensor_dim0_stride=256; no padding/iterate/multicast.
    v8i_ g1 = { 0x20000,            // workgroup_mask=0, data_size=2 (4B)
                0x01000000,         // abar=0 | tensor_dim0[15:0]=256
                0x01000000,         // tensor_dim0[31:16]=0 | tensor_dim1[15:0]=256
                0x01000000,         // tensor_dim1[31:16]=0 | tile_dim0=256
                0x00000100,         // tile_dim1=256 | tile_dim2=0
                0x00000100,         // tensor_dim0_stride[31:0]=256
                0,                  // stride0[47:32]=0 | stride1[15:0]=0
                0 };                // stride1[47:16]=0
    v4i_ gz = { 0, 0, 0, 0 };       // groups 2/3: dims unused (2-D tensor)
#if __clang_major__ >= 23
    v8i_ gz8 = { 0, 0, 0, 0, 0, 0, 0, 0 };
    __builtin_amdgcn_tensor_load_to_lds(g0, g1, gz, gz, gz8, 0);
#else
    __builtin_amdgcn_tensor_load_to_lds(g0, g1, gz, gz, 0);
#endif
    __builtin_amdgcn_s_wait_tensorcnt(0);
  }
  __syncthreads();
#elif defined(USE_ASYNC_LDS)
  // Async DMA-style copy: each lane moves 16B per op; tracked by ASYNCcnt.
  for (int i = tid * 4; i < HW_; i += PLANE_NT * 4) {
    __builtin_amdgcn_global_load_async_to_lds_b128(
        (gas_v4i)(src + i),
        (las_v4i)(plane + i),
        0, 0);
  }
  __builtin_amdgcn_s_wait_asynccnt(0);
  __syncthreads();
#else
  for (int i = tid * 4; i < HW_; i += PLANE_NT * 4) {
    const float4 v = *(const float4*)(src + i);
    *(float4*)(plane + i) = v;
  }
  __syncthreads();
#endif

  // ---- point_sample: bilinear, out (B,N,C) ----
  const float* pc = point_coords + (size_t)b * N_ * 2;
  float* out_samp = out + OFF_SAMP;
  for (int n = tid; n < N_; n += nt) {
    float gx = pc[n * 2 + 0];
    float gy = pc[n * 2 + 1];
    float fx = gx * (float)W_ - 0.5f;
    float fy = gy * (float)H_ - 0.5f;
    float fx0 = floorf(fx), fy0 = floorf(fy);
    int x0 = (int)fx0, y0 = (int)fy0;
    float wx = fx - fx0, wy = fy - fy0;
    int x1 = x0 + 1, y1 = y0 + 1;
    bool xin0 = (x0 >= 0) && (x0 < W_);
    bool xin1 = (x1 >= 0) && (x1 < W_);
    bool yin0 = (y0 >= 0) && (y0 < H_);
    bool yin1 = (y1 >= 0) && (y1 < H_);
    float v00 = (yin0 && xin0) ? plane[y0 * W_ + x0] : 0.f;
    float v01 = (yin0 && xin1) ? plane[y0 * W_ + x1] : 0.f;
    float v10 = (yin1 && xin0) ? plane[y1 * W_ + x0] : 0.f;
    float v11 = (yin1 && xin1) ? plane[y1 * W_ + x1] : 0.f;
    float r = v00 * (1.f - wx) * (1.f - wy) + v01 * wx * (1.f - wy)
            + v10 * (1.f - wx) * wy        + v11 * wx * wy;
    out_samp[((size_t)b * N_ + (size_t)n) * C_ + (size_t)c] = r;
  }

  // ---- extract_local_features: 3x3 valid-mean, out (B,C,N) ----
  const float* coords = out + OFF_COORDS + (size_t)b * TOPK_ * 2;  // written by topk_kernel
  float* out_local = out + OFF_LOCAL;
  for (int j = tid; j < TOPK_; j += nt) {
    float cy = coords[j * 2 + 0];
    float cx = coords[j * 2 + 1];
    int x = (int)(cx * (float)(W_ - 1));  // same float roundtrip as reference
    int y = (int)(cy * (float)(H_ - 1));
    int xmin = (x - 1 < 0) ? 0 : x - 1;
    int xmax = (x + 2 > W_) ? W_ : x + 2;
    int ymin = (y - 1 < 0) ? 0 : y - 1;
    int ymax = (y + 2 > H_) ? H_ : y + 2;
    float s = 0.f; int cnt = 0;
    #pragma unroll
    for (int oy = 0; oy < 3; ++oy) {
      int yy = ymin + oy;
      int yyc = (yy > H_ - 1) ? (H_ - 1) : yy;
      #pragma unroll
      for (int ox = 0; ox < 3; ++ox) {
        int xx = xmin + ox;
        int xxc = (xx > W_ - 1) ? (W_ - 1) : xx;
        if ((yy < ymax) && (xx < xmax)) { s += plane[yyc * W_ + xxc]; ++cnt; }
      }
    }
    out_local[((size_t)b * C_ + (size_t)c) * TOPK_ + (size_t)j] = s / (float)cnt;
  }
}

// ---------------------------------------------------------------------------
extern "C" void kernel_launch(void* const* d_in, const int* in_sizes, int n_in,
                              void* d_out, int out_size, void* d_ws, size_t ws_size,
                              hipStream_t stream) {
  const float* feature_map  = (const float*)d_in[0];
  const float* pred_mask    = (const float*)d_in[1];
  const int*   edge_masks   = (const int*)d_in[2];
  const float* point_coords = (const float*)d_in[3];
  (void)in_sizes; (void)n_in; (void)out_size; (void)d_ws; (void)ws_size;
  float* out = (float*)d_out;

  const size_t shA = (size_t)SELCAP * 8 + 1024 + 64;   // ~66.6 KB
  topk_kernel<<<dim3(B_), dim3(1024), shA, stream>>>(pred_mask, edge_masks, out);

  const size_t shB = (size_t)HW_ * sizeof(float);      // 256 KB plane in LDS
  plane_kernel<<<dim3(C_, B_), dim3(PLANE_NT), shB, stream>>>(feature_map, point_coords, out);
}